// OrderFreeLosses_52982716564246
// MI455X (gfx1250) — compile-verified
//
#include <hip/hip_runtime.h>
#include <cstdint>

// Order-free loss on MI455X (gfx1250, wave32).
// Output scalar = -mean_{s,b} outputs[s, b, tt[s,b]] where tt is the greedy
// sequential argmax over the remaining multi-hot target set (EOS re-armed when
// the set empties). Mask cancels in the reference, so it is not needed.
//
// Plan: one wave32 per batch row.
//   1) async-copy the 32KB targets row into LDS (global_load_async_to_lds_b128)
//   2) ballot-compact indices where targets > 0.5 (ascending -> EOS is slot 0)
//   3) 20 sequential steps: per-lane local max over owned slots (scattered
//      global gathers, ~21 per row per step), wave32 shfl_xor max-reduce with
//      lowest-index tie-break, remove winner, re-arm EOS if the set empties.
//      Prefetch next step's gather addresses (global_prefetch_b8).
//   4) deterministic single-block tree reduction of per-row sums.

#define MAXT 128   // cap on targets per row (~21 expected; P(>128) ~ 0)

__global__ __launch_bounds__(32) void orderfree_row_kernel(
    const float* __restrict__ outputs,   // (S, B, L) log-probs
    const float* __restrict__ targets,   // (B, L) multi-hot
    float* __restrict__ row_loss,        // (B) sum of selected log-probs
    int S, int B, int L)
{
    const int b    = blockIdx.x;
    const int lane = threadIdx.x;

    extern __shared__ float buf[];       // L floats: staged targets row
    __shared__ int ilist[MAXT];          // compacted target indices (ascending)

    const float* row = targets + (size_t)b * (size_t)L;

    // ---- Stage targets row into LDS with CDNA5 async global->LDS DMA ----
    // Each lane moves 16B per op; requires L % 128 == 0 (L = 8192 here).
    {
        const unsigned lds_base = (unsigned)(uintptr_t)buf;  // LDS byte offset
        for (int c = lane * 4; c < L; c += 32 * 4) {
            unsigned lds_addr = lds_base + (unsigned)c * 4u;
            unsigned long long gaddr = (unsigned long long)(uintptr_t)(row + c);
            asm volatile("global_load_async_to_lds_b128 %0, %1, off"
                         :: "v"(lds_addr), "v"(gaddr)
                         : "memory");
        }
        asm volatile("s_wait_asynccnt 0x0" ::: "memory");
    }
    __syncthreads();  // single-wave WG: lowers to s_nop, acts as compiler fence

    // ---- Ballot-compact target indices in ascending order ----
    int cnt = 0;
    for (int base = 0; base < L; base += 32) {
        float t = buf[base + lane];
        bool  p = (t > 0.5f);
        unsigned m = (unsigned)__ballot(p);   // wave32: low 32 bits
        if (p) {
            int pos = cnt + __popc(m & ((1u << lane) - 1u));
            if (pos < MAXT) ilist[pos] = base + lane;
        }
        cnt += __popc(m);
    }
    if (cnt > MAXT) cnt = MAXT;
    __syncthreads();

    // Lane owns slots p = lane + 32k, k = 0..3. EOS (index 0) is slot 0 of
    // lane 0 because targets[:,0] == 1 always and the list is ascending.
    int idx[4];
    unsigned act = 0u;
#pragma unroll
    for (int k = 0; k < 4; ++k) {
        int p = lane + 32 * k;
        if (p < cnt) { idx[k] = ilist[p]; act |= (1u << k); }
        else         { idx[k] = 0x7fffffff; }
    }

    // ---- Sequential greedy argmax over S steps ----
    float acc = 0.0f;
    for (int s = 0; s < S; ++s) {
        const float* o = outputs + ((size_t)s * (size_t)B + (size_t)b) * (size_t)L;

        float best = -__builtin_inff();
        int   bi   = 0x7fffffff;
#pragma unroll
        for (int k = 0; k < 4; ++k) {
            if (act & (1u << k)) {
                float v = o[idx[k]];
                if (v > best || (v == best && idx[k] < bi)) { best = v; bi = idx[k]; }
            }
        }
        // wave32 max-reduce, ties -> smaller index (matches jnp.argmax)
#pragma unroll
        for (int off = 16; off >= 1; off >>= 1) {
            float ov = __shfl_xor(best, off, 32);
            int   oi = __shfl_xor(bi,   off, 32);
            if (ov > best || (ov == best && oi < bi)) { best = ov; bi = oi; }
        }
        acc += best;  // best == outputs[s, b, tt]

        // remove the winner from the active set
#pragma unroll
        for (int k = 0; k < 4; ++k)
            if ((act & (1u << k)) && idx[k] == bi) act &= ~(1u << k);

        // if the set emptied, re-arm EOS (slot 0 of lane 0 holds index 0)
        unsigned long long any = __ballot(act != 0u);
        if (any == 0ull && lane == 0) act |= 1u;

        // prefetch next step's gather addresses into cache
        if (s + 1 < S) {
            const float* on = outputs + ((size_t)(s + 1) * (size_t)B + (size_t)b) * (size_t)L;
#pragma unroll
            for (int k = 0; k < 4; ++k)
                if (act & (1u << k)) __builtin_prefetch(on + idx[k], 0, 1);
        }
    }

    if (lane == 0) row_loss[b] = acc;
}

// Deterministic tree reduction of B per-row sums -> scalar loss.
__global__ __launch_bounds__(256) void orderfree_reduce_kernel(
    const float* __restrict__ row_loss, float* __restrict__ out,
    int B, float scale)
{
    __shared__ float sh[256];
    const int t = threadIdx.x;
    float s = 0.0f;
    for (int i = t; i < B; i += 256) s += row_loss[i];
    sh[t] = s;
    __syncthreads();
#pragma unroll
    for (int off = 128; off >= 1; off >>= 1) {
        if (t < off) sh[t] += sh[t + off];
        __syncthreads();
    }
    if (t == 0) out[0] = -sh[0] * scale;   // -mean over (S,B)
}

extern "C" void kernel_launch(void* const* d_in, const int* in_sizes, int n_in,
                              void* d_out, int out_size, void* d_ws, size_t ws_size,
                              hipStream_t stream)
{
    const float* outputs = (const float*)d_in[0];  // (S,B,L) float32
    const float* targets = (const float*)d_in[1];  // (B,L)   float32
    // d_in[2] = output_symbols (S,B) int32 — unused by the math.

    const int SBL = in_sizes[0];
    const int BL  = in_sizes[1];
    const int SB  = in_sizes[2];
    const int S   = SBL / BL;   // 20
    const int B   = SB / S;     // 512
    const int L   = BL / B;     // 8192

    float* row_loss = (float*)d_ws;       // B floats of scratch
    float* out      = (float*)d_out;      // 1 float

    const size_t shbytes = (size_t)L * sizeof(float);   // 32 KB staged row
    orderfree_row_kernel<<<B, 32, shbytes, stream>>>(outputs, targets, row_loss,
                                                     S, B, L);
    orderfree_reduce_kernel<<<1, 256, 0, stream>>>(row_loss, out, B,
                                                   1.0f / (float)(S * B));
}